// EvolveGCN_3719441678530
// MI455X (gfx1250) — compile-verified
//
#include <hip/hip_runtime.h>
#include <hip/hip_bf16.h>
#include <cfloat>

// ---------------- problem constants (match reference) ----------------
#define TT   16
#define NN   20000
#define F0   128
#define F1   128
#define F2   64
#define NNZ  320000
#define EE   200000

typedef float v2f __attribute__((ext_vector_type(2)));
typedef float v8f __attribute__((ext_vector_type(8)));

// ---------------------------------------------------------------------
// fp32 WMMA 16x16x4 (CDNA5 ISA 7.12.2 layouts).
//   A frag: lane r = l&15 holds A[m][kb], A[m][kb+1], kb = kk*4 + (l>>4)*2
//   B frag: lane r holds B[kb][n], B[kb+1][n]
//   C/D:    VGPR v: M = v + (l>>4)*8, N = l&15
// All GEMM operands are pre-packed into fragment-major float2 streams so
// the hot loops are pure coalesced global_load_b64 / ds_load_b64 + wmma.
// ---------------------------------------------------------------------

// B packed: Bf[(kk*ntiles + tn)*32 + lane] = {B[kb][n], B[kb+1][n]}
__global__ void packB_kernel(const float* __restrict__ B, int ldb, int ntiles,
                             int ktiles, float* __restrict__ Bf) {
  int q = blockIdx.x * blockDim.x + threadIdx.x;
  if (q >= ktiles * ntiles * 32) return;
  int lane = q & 31, tile = q >> 5;
  int kk = tile / ntiles, tn = tile % ntiles;
  int r = lane & 15, hi = (lane >> 4) * 2;
  int kb = kk * 4 + hi, n = tn * 16 + r;
  ((float2*)Bf)[q] = make_float2(B[kb * ldb + n], B[(kb + 1) * ldb + n]);
}

// A packed: Af[(tm*ktiles + kk)*32 + lane] = {A[m][kb], A[m][kb+1]}
__global__ void packA_kernel(const float* __restrict__ A, int lda, int mtiles,
                             int ktiles, float* __restrict__ Af) {
  int q = blockIdx.x * blockDim.x + threadIdx.x;
  if (q >= mtiles * ktiles * 32) return;
  int lane = q & 31, tile = q >> 5;
  int tm = tile / ktiles, kk = tile % ktiles;
  int m = tm * 16 + (lane & 15), kb = kk * 4 + ((lane >> 4) * 2);
  ((float2*)Af)[q] = make_float2(A[m * lda + kb], A[m * lda + kb + 1]);
}

// C packed: Cf[tile*256 + lane*8 + v] = C[m][n] (two b128 loads per tile)
__global__ void packC_kernel(const float* __restrict__ C, int ldc, int mtiles,
                             int ntiles, float* __restrict__ Cf) {
  int q = blockIdx.x * blockDim.x + threadIdx.x;
  if (q >= mtiles * ntiles * 256) return;
  int v = q & 7, lane = (q >> 3) & 31, tile = q >> 8;
  int tm = tile / ntiles, tn = tile % ntiles;
  int m = tm * 16 + v + ((lane >> 4) * 8), n = tn * 16 + (lane & 15);
  Cf[q] = C[m * ldc + n];
}

// acc += A_packed * B_packed for one (tm,tn) tile, K = 4*ktiles
__device__ __forceinline__ v8f wmma_ff(const float2* __restrict__ Af, int ktiles, int tm,
                                       const float2* __restrict__ Bf, int ntiles, int tn,
                                       v8f acc, int lane) {
  for (int kk = 0; kk < ktiles; ++kk) {
    float2 aa = Af[(tm * ktiles + kk) * 32 + lane];
    float2 bb = Bf[(kk * ntiles + tn) * 32 + lane];
    v2f a = { aa.x, aa.y };
    v2f b = { bb.x, bb.y };
    acc = __builtin_amdgcn_wmma_f32_16x16x4_f32(false, a, false, b,
                                                (short)0, acc, false, false);
  }
  return acc;
}

// ------------------------- p / ||p|| ---------------------------------
__global__ void pnorm_kernel(const float* __restrict__ p, float* __restrict__ pn) {
  __shared__ float s[F0];
  int t = threadIdx.x;
  float v = p[t];
  s[t] = v * v;
  __syncthreads();
  for (int st = F0 / 2; st > 0; st >>= 1) {
    if (t < st) s[t] += s[t + st];
    __syncthreads();
  }
  pn[t] = v / sqrtf(s[0]);
}

// ----------------- y[i] = dot(X[i], p_norm), wave per 8 rows ---------
__global__ void score_kernel(const float* __restrict__ X,
                             const float* __restrict__ pn,
                             float* __restrict__ y) {
  const int wave = blockIdx.x * (blockDim.x >> 5) + (threadIdx.x >> 5);
  const int lane = threadIdx.x & 31;
  const int row0 = wave * 8;
  if (row0 >= TT * NN) return;
  float4 pv = ((const float4*)pn)[lane];
  for (int i = 0; i < 8; ++i) {
    int row = row0 + i;
    if (row >= TT * NN) break;                       // wave-uniform
    float4 xv = ((const float4*)(X + (size_t)row * F0))[lane];
    float d = xv.x * pv.x + xv.y * pv.y + xv.z * pv.z + xv.w * pv.w;
    for (int off = 16; off > 0; off >>= 1) d += __shfl_xor(d, off);
    if (lane == 0) y[row] = d;
  }
}

// ---------------- top-k: two-phase bitonic selection -----------------
// rank order: val descending, index ascending on ties (matches jax top_k)
__device__ __forceinline__ bool ranks_before(float av, int ai, float bv, int bi) {
  return (av > bv) || (av == bv && ai < bi);
}

#define TOPK_BLOCKS 40
#define TOPK_CHUNK  500

__global__ void topk_phase1(const float* __restrict__ yt,
                            float* __restrict__ cv, int* __restrict__ ci) {
  __shared__ float v[512];
  __shared__ int   ix[512];
  const int tid = threadIdx.x;                       // 512 threads
  const int base = blockIdx.x * TOPK_CHUNK;
  if (tid < TOPK_CHUNK) { v[tid] = yt[base + tid]; ix[tid] = base + tid; }
  else                  { v[tid] = -FLT_MAX;        ix[tid] = 0x7fffffff; }
  __syncthreads();
  for (int k = 2; k <= 512; k <<= 1) {
    for (int j = k >> 1; j > 0; j >>= 1) {
      int i = tid, p = i ^ j;
      if (p > i) {
        bool keep = ((i & k) == 0);
        bool bef = ranks_before(v[i], ix[i], v[p], ix[p]);
        if (bef != keep) {
          float tv = v[i]; v[i] = v[p]; v[p] = tv;
          int ti = ix[i]; ix[i] = ix[p]; ix[p] = ti;
        }
      }
      __syncthreads();
    }
  }
  if (tid < F1) { cv[blockIdx.x * F1 + tid] = v[tid]; ci[blockIdx.x * F1 + tid] = ix[tid]; }
}

// merge 40*128 = 5120 candidates (pad 8192), take top-128, build Xs.
__global__ void topk_phase2(const float* __restrict__ cv, const int* __restrict__ ci,
                            const float* __restrict__ Xt, float* __restrict__ Xs) {
  extern __shared__ float lds[];
  float* v = lds;                 // 8192
  int*   ix = (int*)(lds + 8192); // 8192
  const int tid = threadIdx.x;    // 1024 threads
  const int NCAND = TOPK_BLOCKS * F1;
  for (int i = tid; i < 8192; i += 1024) {
    if (i < NCAND) { v[i] = cv[i]; ix[i] = ci[i]; }
    else           { v[i] = -FLT_MAX; ix[i] = 0x7fffffff; }
  }
  __syncthreads();
  for (int k = 2; k <= 8192; k <<= 1) {
    for (int j = k >> 1; j > 0; j >>= 1) {
      for (int i = tid; i < 8192; i += 1024) {
        int p = i ^ j;
        if (p > i) {
          bool keep = ((i & k) == 0);
          bool bef = ranks_before(v[i], ix[i], v[p], ix[p]);
          if (bef != keep) {
            float tv = v[i]; v[i] = v[p]; v[p] = tv;
            int ti = ix[i]; ix[i] = ix[p]; ix[p] = ti;
          }
        }
      }
      __syncthreads();
    }
  }
  // Xs[f][j] = X_t[idx[j]][f] * vals[j]   (top-128 now in v/ix[0..127])
  for (int e = tid; e < F0 * F1; e += 1024) {
    int f = e >> 7, j = e & 127;
    Xs[f * F1 + j] = Xt[(size_t)ix[j] * F0 + f] * v[j];
  }
}

// -------------------- GRU stage 1: Z and R*W -------------------------
// 8 blocks x 8 waves = 64 waves; wave w computes tile (tm,tn)=(w>>3,w&7).
//   Z  = sigmoid(Wz@Xs + Uz@W + Bz)   -> Zbuf
//   RW = sigmoid(Wr@Xs + Ur@W + Br)*W -> RWbuf
__global__ __launch_bounds__(256, 1)
void zr_kernel(const float* __restrict__ XsF, const float* __restrict__ WF,
               const float* __restrict__ W,
               const float* __restrict__ WzF, const float* __restrict__ UzF,
               const float* __restrict__ BzF,
               const float* __restrict__ WrF, const float* __restrict__ UrF,
               const float* __restrict__ BrF,
               float* __restrict__ Zbuf, float* __restrict__ RWbuf) {
  const int lane = threadIdx.x & 31;
  const int wave = blockIdx.x * 8 + (threadIdx.x >> 5);   // == tile id, 0..63
  const int tm = wave >> 3, tn = wave & 7;
  const int r = lane & 15, half = lane >> 4;
  v8f z, rr;
  for (int v = 0; v < 8; ++v) {
    z[v]  = BzF[wave * 256 + lane * 8 + v];
    rr[v] = BrF[wave * 256 + lane * 8 + v];
  }
  z  = wmma_ff((const float2*)WzF, 32, tm, (const float2*)XsF, 8, tn, z,  lane);
  z  = wmma_ff((const float2*)UzF, 32, tm, (const float2*)WF,  8, tn, z,  lane);
  rr = wmma_ff((const float2*)WrF, 32, tm, (const float2*)XsF, 8, tn, rr, lane);
  rr = wmma_ff((const float2*)UrF, 32, tm, (const float2*)WF,  8, tn, rr, lane);
  for (int v = 0; v < 8; ++v) {
    int m = tm * 16 + v + half * 8, n = tn * 16 + r;
    float zg = 1.0f / (1.0f + expf(-z[v]));
    float rg = 1.0f / (1.0f + expf(-rr[v]));
    Zbuf[m * F1 + n]  = zg;
    RWbuf[m * F1 + n] = rg * W[m * F1 + n];
  }
}

// ------------- GRU stage 2: Ht and in-place W blend ------------------
//   Ht = tanh(Wh@Xs + Uh@(R*W) + Bh);  W = (1-Z)*W + Z*Ht (elementwise)
__global__ __launch_bounds__(256, 1)
void h_kernel(const float* __restrict__ XsF, const float* __restrict__ RWF,
              const float* __restrict__ WhF, const float* __restrict__ UhF,
              const float* __restrict__ BhF,
              const float* __restrict__ Zbuf, float* __restrict__ W) {
  const int lane = threadIdx.x & 31;
  const int wave = blockIdx.x * 8 + (threadIdx.x >> 5);
  const int tm = wave >> 3, tn = wave & 7;
  const int r = lane & 15, half = lane >> 4;
  v8f h;
  for (int v = 0; v < 8; ++v) h[v] = BhF[wave * 256 + lane * 8 + v];
  h = wmma_ff((const float2*)WhF, 32, tm, (const float2*)XsF, 8, tn, h, lane);
  h = wmma_ff((const float2*)UhF, 32, tm, (const float2*)RWF, 8, tn, h, lane);
  for (int v = 0; v < 8; ++v) {
    int m = tm * 16 + v + half * 8, n = tn * 16 + r;
    float ht = tanhf(h[v]);
    float zg = Zbuf[m * F1 + n];
    float w  = W[m * F1 + n];
    W[m * F1 + n] = (1.0f - zg) * w + zg * ht;
  }
}

// ----------------------------- zero ----------------------------------
__global__ void zero_kernel(float* __restrict__ buf, int n) {
  int i = blockIdx.x * blockDim.x + threadIdx.x;
  if (i < n) buf[i] = 0.0f;
}

// ------------- agg[trg] += val * X_t[src]  (COO scatter) -------------
__global__ void spmm_kernel(const int* __restrict__ Aidx,   // (NNZ,2): [trg,src]
                            const float* __restrict__ Aval,
                            const float* __restrict__ Xt,
                            float* __restrict__ agg) {
  int gid = blockIdx.x * blockDim.x + threadIdx.x;
  int e = gid >> 5, c = gid & 31;
  if (e >= NNZ) return;
  int trg = Aidx[e * 2 + 0], src = Aidx[e * 2 + 1];
  float val = Aval[e];
  float4 x = ((const float4*)(Xt + (size_t)src * F0))[c];
  float* a = agg + (size_t)trg * F0 + c * 4;
  atomicAdd(a + 0, val * x.x);
  atomicAdd(a + 1, val * x.y);
  atomicAdd(a + 2, val * x.z);
  atomicAdd(a + 3, val * x.w);
}

// ---------------- h_t = agg (N x 128) @ W (128 x 128) ----------------
// Wave per 16-row strip; A staged in LDS (pitch 132 -> conflict-free),
// B from packed fragments (coalesced b64, L2-broadcast).
#define AP 132
__global__ __launch_bounds__(256, 1)
void gemm_h_kernel(const float* __restrict__ agg,
                   const float* __restrict__ Wf,
                   float* __restrict__ Yt) {
  extern __shared__ float As[];                 // 8 waves * 16*AP floats
  const int wave = threadIdx.x >> 5, lane = threadIdx.x & 31;
  const int gw = blockIdx.x * 8 + wave;
  const bool active = gw < NN / 16;
  float* Aw = As + wave * 16 * AP;
  if (active) {
    const float* Ag = agg + (size_t)gw * 16 * F0;
    for (int i = 0; i < 16; ++i)
      ((float4*)(Aw + i * AP))[lane] = ((const float4*)(Ag + i * F0))[lane];
  }
  __syncthreads();
  if (!active) return;
  const int r = lane & 15, hi = (lane >> 4) * 2;
  const float2* Bf = (const float2*)Wf;
  v8f acc[8];
  for (int tn = 0; tn < 8; ++tn) acc[tn] = (v8f){};
  for (int kk = 0; kk < F0 / 4; ++kk) {
    const int kb = kk * 4 + hi;
    v2f a = { Aw[r * AP + kb], Aw[r * AP + kb + 1] };
    for (int tn = 0; tn < 8; ++tn) {
      float2 bb = Bf[(kk * 8 + tn) * 32 + lane];
      v2f b = { bb.x, bb.y };
      acc[tn] = __builtin_amdgcn_wmma_f32_16x16x4_f32(false, a, false, b,
                                                      (short)0, acc[tn], false, false);
    }
  }
  for (int tn = 0; tn < 8; ++tn)
    for (int v = 0; v < 8; ++v) {
      int m = gw * 16 + v + (lane >> 4) * 8;
      int n = tn * 16 + r;
      Yt[(size_t)m * F1 + n] = acc[tn][v];
    }
}

// ------- out = [Y[lin_src], Y[lin_trg]] (E x 256) @ U (256 x 64) -----
// Wave per 16 edges; concatenated 256-float rows staged contiguously in
// LDS (pitch 260 -> conflict-free), U from packed fragments.
#define EP 260
__global__ __launch_bounds__(256, 1)
void edge_gemm_kernel(const float* __restrict__ Y,
                      const int* __restrict__ edges,   // (3,E)
                      const float* __restrict__ Uf,
                      float* __restrict__ out) {
  extern __shared__ float Ls[];                 // 8 waves * 16*EP floats
  const int wave = threadIdx.x >> 5, lane = threadIdx.x & 31;
  const int gw = blockIdx.x * 8 + wave;
  const bool active = gw < EE / 16;
  float* Lw = Ls + wave * 16 * EP;
  if (active) {
    for (int i = 0; i < 16; ++i) {
      int e = gw * 16 + i;
      int t = edges[e], s = edges[EE + e], g = edges[2 * EE + e];
      const float* rs = Y + ((size_t)t * NN + s) * F1;
      const float* rt = Y + ((size_t)t * NN + g) * F1;
      ((float4*)(Lw + i * EP))[lane]       = ((const float4*)rs)[lane];
      ((float4*)(Lw + i * EP + F1))[lane]  = ((const float4*)rt)[lane];
    }
  }
  __syncthreads();
  if (!active) return;
  const int r = lane & 15, hi = (lane >> 4) * 2;
  const float2* Bf = (const float2*)Uf;
  v8f acc[4];
  for (int tn = 0; tn < 4; ++tn) acc[tn] = (v8f){};
  for (int kk = 0; kk < (2 * F1) / 4; ++kk) {
    const int kb = kk * 4 + hi;
    v2f a = { Lw[r * EP + kb], Lw[r * EP + kb + 1] };
    for (int tn = 0; tn < 4; ++tn) {
      float2 bb = Bf[(kk * 4 + tn) * 32 + lane];
      v2f b = { bb.x, bb.y };
      acc[tn] = __builtin_amdgcn_wmma_f32_16x16x4_f32(false, a, false, b,
                                                      (short)0, acc[tn], false, false);
    }
  }
  for (int tn = 0; tn < 4; ++tn)
    for (int v = 0; v < 8; ++v) {
      int m = gw * 16 + v + (lane >> 4) * 8;
      int n = tn * 16 + r;
      out[(size_t)m * F2 + n] = acc[tn][v];
    }
}

// ---------------------------------------------------------------------
extern "C" void kernel_launch(void* const* d_in, const int* in_sizes, int n_in,
                              void* d_out, int out_size, void* d_ws, size_t ws_size,
                              hipStream_t stream) {
  const float* X      = (const float*)d_in[0];
  const float* A_val  = (const float*)d_in[1];
  const int*   A_idx  = (const int*)  d_in[2];
  const int*   edges  = (const int*)  d_in[3];
  const float* p      = (const float*)d_in[4];
  const float* Wz     = (const float*)d_in[5];
  const float* Uz     = (const float*)d_in[6];
  const float* Bz     = (const float*)d_in[7];
  const float* Wr     = (const float*)d_in[8];
  const float* Ur     = (const float*)d_in[9];
  const float* Br     = (const float*)d_in[10];
  const float* Wh     = (const float*)d_in[11];
  const float* Uh     = (const float*)d_in[12];
  const float* Bh     = (const float*)d_in[13];
  const float* W_init = (const float*)d_in[14];
  const float* U      = (const float*)d_in[15];
  float* out = (float*)d_out;

  // workspace layout (floats): ~169 MB total
  const int M = F0 * F1;                    // 16384
  float* ws   = (float*)d_ws;
  float* pn   = ws;                         // 128
  float* W    = pn + F0;                    // GRU state
  float* Xs   = W + M;
  float* XsF  = Xs + M;                     // Xs B-fragments
  float* WF   = XsF + M;                    // W  B-fragments (pre-update)
  float* RWF  = WF + M;                     // R*W B-fragments
  float* Wf   = RWF + M;                    // W  B-fragments (post-update, for gemm_h)
  float* Uf   = Wf + M;                     // U packed (256x64)
  float* Zbuf = Uf + M;
  float* RWbuf= Zbuf + M;
  float* WzF  = RWbuf + M;                  // packed A matrices (6x)
  float* UzF  = WzF + M;
  float* WrF  = UzF + M;
  float* UrF  = WrF + M;
  float* WhF  = UrF + M;
  float* UhF  = WhF + M;
  float* BzF  = UhF + M;                    // packed C biases (3x)
  float* BrF  = BzF + M;
  float* BhF  = BrF + M;
  float* y    = BhF + M;                    // T*N = 320000
  float* cv   = y + TT * NN;                // 5120
  int*   ci   = (int*)(cv + TOPK_BLOCKS * F1);
  float* agg  = cv + 2 * TOPK_BLOCKS * F1;  // N*F0 = 2,560,000
  float* Ybuf = agg + (size_t)NN * F0;      // T*N*F1 = 40,960,000

  hipMemcpyAsync(W, W_init, sizeof(float) * M, hipMemcpyDeviceToDevice, stream);

  pnorm_kernel<<<1, F0, 0, stream>>>(p, pn);
  score_kernel<<<(TT * NN / 8 + 7) / 8, 256, 0, stream>>>(X, pn, y);

  // one-time packs: U (K=256 -> 64 kt, N=64 -> 4 nt), six A mats, three biases
  packB_kernel<<<32, 256, 0, stream>>>(U, F2, 4, 64, Uf);
  packA_kernel<<<32, 256, 0, stream>>>(Wz, F0, 8, 32, WzF);
  packA_kernel<<<32, 256, 0, stream>>>(Uz, F0, 8, 32, UzF);
  packA_kernel<<<32, 256, 0, stream>>>(Wr, F0, 8, 32, WrF);
  packA_kernel<<<32, 256, 0, stream>>>(Ur, F0, 8, 32, UrF);
  packA_kernel<<<32, 256, 0, stream>>>(Wh, F0, 8, 32, WhF);
  packA_kernel<<<32, 256, 0, stream>>>(Uh, F0, 8, 32, UhF);
  packC_kernel<<<64, 256, 0, stream>>>(Bz, F1, 8, 8, BzF);
  packC_kernel<<<64, 256, 0, stream>>>(Br, F1, 8, 8, BrF);
  packC_kernel<<<64, 256, 0, stream>>>(Bh, F1, 8, 8, BhF);

  const size_t p2Lds = 8192 * 8;                          // 64 KB
  const size_t ghLds = sizeof(float) * 8 * 16 * AP;       // ~66 KB
  const size_t egLds = sizeof(float) * 8 * 16 * EP;       // ~130 KB

  for (int t = 0; t < TT; ++t) {
    const float* Xt = X + (size_t)t * NN * F0;
    topk_phase1<<<TOPK_BLOCKS, 512, 0, stream>>>(y + (size_t)t * NN, cv, ci);
    topk_phase2<<<1, 1024, p2Lds, stream>>>(cv, ci, Xt, Xs);
    packB_kernel<<<32, 256, 0, stream>>>(Xs, F1, 8, 32, XsF);
    packB_kernel<<<32, 256, 0, stream>>>(W, F1, 8, 32, WF);
    zr_kernel<<<8, 256, 0, stream>>>(XsF, WF, W, WzF, UzF, BzF, WrF, UrF, BrF,
                                     Zbuf, RWbuf);
    packB_kernel<<<32, 256, 0, stream>>>(RWbuf, F1, 8, 32, RWF);
    h_kernel<<<8, 256, 0, stream>>>(XsF, RWF, WhF, UhF, BhF, Zbuf, W);
    packB_kernel<<<32, 256, 0, stream>>>(W, F1, 8, 32, Wf);
    zero_kernel<<<(NN * F0 + 255) / 256, 256, 0, stream>>>(agg, NN * F0);
    spmm_kernel<<<(NNZ * 32 + 255) / 256, 256, 0, stream>>>(
        A_idx + (size_t)t * NNZ * 2, A_val + (size_t)t * NNZ, Xt, agg);
    gemm_h_kernel<<<(NN / 16 + 7) / 8, 256, ghLds, stream>>>(
        agg, Wf, Ybuf + (size_t)t * NN * F1);
  }
  edge_gemm_kernel<<<(EE / 16 + 7) / 8, 256, egLds, stream>>>(Ybuf, edges, Uf, out);
}